// LightGCN_30554397343958
// MI455X (gfx1250) — compile-verified
//
#include <hip/hip_runtime.h>

// ---------------------------------------------------------------------------
// LightGCN propagation on MI455X (gfx1250).
//
// Roofline: 384 MFLOP vs ~1.5 GB of gather/scatter traffic -> entirely
// memory/atomic bound (AI ~ 0.24 flop/byte), so WMMA has no role (unstructured
// COO, avg degree ~3.3 -> no dense 16x16 tiles). The two 76.8 MB feature
// ping-pong buffers fit in the 192 MB L2, so gathers and f32 atomics resolve
// at L2 speed. gfx1250-specific paths used: async global->LDS edge staging
// (ASYNCcnt + s_wait_asynccnt) and no-return global_atomic_add_f32 scatter.
// ---------------------------------------------------------------------------

#define NFEAT       64
#define TILE_EDGES  1024   // 12 KB LDS per block (rows+cols+vals)
#define BLOCK       256    // 8 wave32 per block

#if defined(__has_builtin)
#if __has_builtin(__builtin_amdgcn_global_load_async_to_lds_b32)
#define HAVE_ASYNC_LDS 1
#endif
#if __has_builtin(__builtin_amdgcn_s_wait_asynccnt)
#define HAVE_WAIT_ASYNC 1
#endif
#endif

typedef __attribute__((address_space(1))) int GasInt;
typedef __attribute__((address_space(3))) int LasInt;

__device__ __forceinline__ void async_copy_b32(const void* gsrc, void* ldst) {
#if defined(HAVE_ASYNC_LDS)
  // (global src, lds dst, imm offset, imm cpol)
  __builtin_amdgcn_global_load_async_to_lds_b32((GasInt*)gsrc, (LasInt*)ldst, 0, 0);
#else
  *(int*)ldst = *(const int*)gsrc;  // synchronous fallback
#endif
}

__device__ __forceinline__ void wait_async_all() {
#if defined(HAVE_ASYNC_LDS)
#if defined(HAVE_WAIT_ASYNC)
  __builtin_amdgcn_s_wait_asynccnt(0);
#else
  asm volatile("s_wait_asynccnt 0" ::: "memory");
#endif
#endif
}

// Guaranteed single-instruction f32 atomic add (no-return form).
__device__ __forceinline__ void atomic_add_f32(float* p, float v) {
  asm volatile("global_atomic_add_f32 %0, %1, off"
               :
               : "v"(p), "v"(v)
               : "memory");
}

// out[r] += v * feat[c] for each COO edge (r, c, v) in this block's tile.
// Edge tile staged to LDS with async global->LDS loads; 16 lanes per edge,
// each lane owns 4 contiguous features (coalesced 256B row gather).
__global__ void __launch_bounds__(BLOCK)
lgcn_spmm_kernel(const int* __restrict__ rows, const int* __restrict__ cols,
                 const float* __restrict__ vals, const float* __restrict__ feat,
                 float* __restrict__ out, int num_edges) {
  __shared__ int   s_rows[TILE_EDGES];
  __shared__ int   s_cols[TILE_EDGES];
  __shared__ float s_vals[TILE_EDGES];

  const int tid  = threadIdx.x;
  const int base = blockIdx.x * TILE_EDGES;
  const int n    = min(TILE_EDGES, num_edges - base);

  // Stage edge tile into LDS (async, ASYNCcnt-tracked).
  for (int i = tid; i < n; i += BLOCK) {
    async_copy_b32(rows + base + i, s_rows + i);
    async_copy_b32(cols + base + i, s_cols + i);
    async_copy_b32(vals + base + i, s_vals + i);
  }
  wait_async_all();
  __syncthreads();

  // Consume: LDS broadcast of (r,c,v) across each 16-lane group.
  const int group = tid >> 4;   // 16 edge-groups per block
  const int lane  = tid & 15;   // feature sub-chunk within the group
  for (int e = group; e < n; e += (BLOCK / 16)) {
    const int   c = s_cols[e];
    const int   r = s_rows[e];
    const float v = s_vals[e];
    const float4 f =
        *(const float4*)(feat + (size_t)c * NFEAT + (size_t)lane * 4);
    float* dst = out + (size_t)r * NFEAT + (size_t)lane * 4;
    atomic_add_f32(dst + 0, v * f.x);
    atomic_add_f32(dst + 1, v * f.y);
    atomic_add_f32(dst + 2, v * f.z);
    atomic_add_f32(dst + 3, v * f.w);
  }
}

// cur = [user_emb ; item_emb], acc = same
__global__ void __launch_bounds__(BLOCK)
lgcn_init_kernel(const float4* __restrict__ ue, const float4* __restrict__ ie,
                 float4* __restrict__ cur, float4* __restrict__ acc,
                 int total4, int user4) {
  for (int i = blockIdx.x * blockDim.x + threadIdx.x; i < total4;
       i += gridDim.x * blockDim.x) {
    float4 v = (i < user4) ? ue[i] : ie[i - user4];
    cur[i] = v;
    acc[i] = v;
  }
}

__global__ void __launch_bounds__(BLOCK)
lgcn_zero_kernel(float4* __restrict__ p, int total4) {
  const float4 z = make_float4(0.f, 0.f, 0.f, 0.f);
  for (int i = blockIdx.x * blockDim.x + threadIdx.x; i < total4;
       i += gridDim.x * blockDim.x)
    p[i] = z;
}

// acc += nxt, and zero zbuf (the buffer the next layer scatters into).
__global__ void __launch_bounds__(BLOCK)
lgcn_acc_zero_kernel(float4* __restrict__ acc, const float4* __restrict__ nxt,
                     float4* __restrict__ zbuf, int total4) {
  const float4 z = make_float4(0.f, 0.f, 0.f, 0.f);
  for (int i = blockIdx.x * blockDim.x + threadIdx.x; i < total4;
       i += gridDim.x * blockDim.x) {
    float4 a = acc[i];
    float4 b = nxt[i];
    a.x += b.x; a.y += b.y; a.z += b.z; a.w += b.w;
    acc[i]  = a;
    zbuf[i] = z;
  }
}

__global__ void __launch_bounds__(BLOCK)
lgcn_acc_kernel(float4* __restrict__ acc, const float4* __restrict__ nxt,
                int total4) {
  for (int i = blockIdx.x * blockDim.x + threadIdx.x; i < total4;
       i += gridDim.x * blockDim.x) {
    float4 a = acc[i];
    float4 b = nxt[i];
    a.x += b.x; a.y += b.y; a.z += b.z; a.w += b.w;
    acc[i] = a;
  }
}

extern "C" void kernel_launch(void* const* d_in, const int* in_sizes, int n_in,
                              void* d_out, int out_size, void* d_ws,
                              size_t ws_size, hipStream_t stream) {
  const float* user_emb = (const float*)d_in[0];
  const float* item_emb = (const float*)d_in[1];
  const int*   rows     = (const int*)d_in[2];
  const int*   cols     = (const int*)d_in[3];
  const float* vals     = (const float*)d_in[4];

  const int user_elems = in_sizes[0];           // NUM_USERS * 64
  const int item_elems = in_sizes[1];           // NUM_ITEMS * 64
  const int num_edges  = in_sizes[2];
  const int total      = user_elems + item_elems;  // N * 64 == out_size
  const int total4     = total / 4;
  const int user4      = user_elems / 4;

  float* bufA = (float*)d_ws;                   // ping
  float* bufB = bufA + total;                   // pong (needs 2*N*64*4 bytes)
  float* acc  = (float*)d_out;

  const int gb_elem = (total4 + BLOCK - 1) / BLOCK;
  const int gb_spmm = (num_edges + TILE_EDGES - 1) / TILE_EDGES;

  // acc = cur = concat(user_emb, item_emb); bufB = 0 (layer-0 scatter target)
  lgcn_init_kernel<<<gb_elem, BLOCK, 0, stream>>>(
      (const float4*)user_emb, (const float4*)item_emb, (float4*)bufA,
      (float4*)acc, total4, user4);
  lgcn_zero_kernel<<<gb_elem, BLOCK, 0, stream>>>((float4*)bufB, total4);

  // Layer 0: bufA -> bufB ; acc += bufB ; zero bufA (layer-1 target)
  lgcn_spmm_kernel<<<gb_spmm, BLOCK, 0, stream>>>(rows, cols, vals, bufA, bufB,
                                                  num_edges);
  lgcn_acc_zero_kernel<<<gb_elem, BLOCK, 0, stream>>>(
      (float4*)acc, (const float4*)bufB, (float4*)bufA, total4);

  // Layer 1: bufB -> bufA ; acc += bufA ; zero bufB (layer-2 target)
  lgcn_spmm_kernel<<<gb_spmm, BLOCK, 0, stream>>>(rows, cols, vals, bufB, bufA,
                                                  num_edges);
  lgcn_acc_zero_kernel<<<gb_elem, BLOCK, 0, stream>>>(
      (float4*)acc, (const float4*)bufA, (float4*)bufB, total4);

  // Layer 2: bufA -> bufB ; acc += bufB
  lgcn_spmm_kernel<<<gb_spmm, BLOCK, 0, stream>>>(rows, cols, vals, bufA, bufB,
                                                  num_edges);
  lgcn_acc_kernel<<<gb_elem, BLOCK, 0, stream>>>((float4*)acc,
                                                 (const float4*)bufB, total4);
}